// CannyFilter_32384053412244
// MI455X (gfx1250) — compile-verified
//
#include <hip/hip_runtime.h>
#include <math.h>
#include <stdint.h>

// Fused Canny edge filter for MI455X (gfx1250).
// Bandwidth-bound (~101 MB -> ~4.3 us at 23.3 TB/s). One pass:
//   async global->LDS tile staging (ASYNCcnt path),
//   horizontal gaussian as banded matmul on the WMMA pipe,
//   vertical gaussian + Sobel + hypot + NMS in VALU/LDS.

typedef __attribute__((ext_vector_type(2))) float v2f;
typedef __attribute__((ext_vector_type(8))) float v8f;

namespace {
constexpr int HH = 512, WW = 512;
constexpr int TILE = 32;
constexpr int IR = TILE + 8;     // 40: img tile rows/cols (halo 4)
constexpr int IPITCH = 44;       // img LDS pitch (multiple of 4 for b128 alignment)
constexpr int HROWS = 48;        // hblur LDS rows (3x 16-row WMMA blocks)
constexpr int HPITCH = 49;       // hblur LDS pitch (3x16 cols + pad)
constexpr int BR = TILE + 4;     // 36: blurred rows/cols (halo 2)
constexpr int BPITCH = BR + 1;   // 37
constexpr int GR = TILE + 2;     // 34: grad-mag rows/cols (halo 1)
constexpr int GPITCH = GR + 1;   // 35
} // namespace

__device__ __forceinline__ int imin(int a, int b) { return a < b ? a : b; }

__device__ __forceinline__ float band_tap(int d, float g0, float g1, float g2,
                                          float g3, float g4) {
    float v = 0.0f;
    v = (d == 0) ? g0 : v;
    v = (d == 1) ? g1 : v;
    v = (d == 2) ? g2 : v;
    v = (d == 3) ? g3 : v;
    v = (d == 4) ? g4 : v;
    return v;
}

__global__ __launch_bounds__(256)
void canny_fused_kernel(const float* __restrict__ img,
                        const float* __restrict__ wg,
                        float* __restrict__ out)
{
    __shared__ float s_img[IR * IPITCH];      // raw image tile (zero outside image)
    __shared__ float s_hb [HROWS * HPITCH];   // horizontal gaussian (WMMA output)
    __shared__ float s_bl [BR * BPITCH];      // full gaussian (zero outside image)
    __shared__ float s_gm [GR * GPITCH];      // gradient magnitude (zero outside image)

    const int tid = threadIdx.x;
    const int x0 = blockIdx.x * TILE;
    const int y0 = blockIdx.y * TILE;
    const long long base = (long long)blockIdx.z * (HH * WW);

    // Gaussian taps (unnormalized exp(-0.5*k^2)) read from the reference weights.
    const float g0 = wg[0], g1 = wg[1], g2 = wg[2], g3 = wg[3], g4 = wg[4];

    // ---- Stage A: async-load 40x40 img tile (halo 4) into LDS ----------------
    // 40 rows x 10 b128 groups = 400 transfers. The x-halo of 4 floats means
    // every 4-float group is either fully inside or fully outside the image.
    const unsigned lds_img_base = (unsigned)(uintptr_t)(void*)s_img;
    #pragma unroll
    for (int k = 0; k < 2; ++k) {
        const int i = tid + k * 256;
        if (i < IR * 10) {
            const int r = i / 10;
            const int j = i - r * 10;
            const int gy = y0 - 4 + r;
            const int gxs = x0 - 4 + 4 * j;            // 16B-aligned global col
            const int lidx = r * IPITCH + 4 * j;       // 16B-aligned LDS offset
            if ((unsigned)gy < (unsigned)HH && (unsigned)gxs < (unsigned)WW) {
                const unsigned lds_addr = lds_img_base + 4u * (unsigned)lidx;
                const unsigned long long gaddr =
                    (unsigned long long)(uintptr_t)(img + base + (long long)gy * WW + gxs);
                asm volatile("global_load_async_to_lds_b128 %0, %1, off"
                             :: "v"(lds_addr), "v"(gaddr) : "memory");
            } else {
                *(float4*)&s_img[lidx] = make_float4(0.f, 0.f, 0.f, 0.f);
            }
        }
    }
    asm volatile("s_wait_asynccnt 0x0" ::: "memory");
    __syncthreads();

    // ---- Stage B: horizontal gaussian as banded matmul on the WMMA pipe ------
    // hblur(y, j) = sum_x img(y, j0+x) * W(x, j-j0), W(x,j) = g[x-j] (20x16 band),
    // done as 5 chained V_WMMA_F32_16X16X4_F32 per 16x16 block; 3x3 blocks cover
    // the 40x36 region (pad rows/cols produce clamped garbage never consumed).
    {
        const int lane = tid & 31;
        const int wave = tid >> 5;
        const int nlo  = lane & 15;             // M (A) / N (B,D) index
        const int kkb  = (lane >> 4) << 1;      // K sub-index base: 0 or 2

        // Constant band-weight B fragments per K-chunk (no LDS traffic).
        v2f bfrag[5];
        #pragma unroll
        for (int q = 0; q < 5; ++q) {
            const int xx = 4 * q + kkb;
            bfrag[q][0] = band_tap(xx - nlo,     g0, g1, g2, g3, g4);
            bfrag[q][1] = band_tap(xx + 1 - nlo, g0, g1, g2, g3, g4);
        }

        for (int t = wave; t < 9; t += 8) {     // wave-uniform => EXEC all ones
            const int br = t / 3, bc = t - 3 * br;
            const int row = imin(br * 16 + nlo, IR - 1);   // clamp into init'd LDS
            v8f acc = {0.f, 0.f, 0.f, 0.f, 0.f, 0.f, 0.f, 0.f};
            #pragma unroll
            for (int q = 0; q < 5; ++q) {
                const int cb = bc * 16 + 4 * q + kkb;
                v2f a;
                a[0] = s_img[row * IPITCH + imin(cb,     IR - 1)];
                a[1] = s_img[row * IPITCH + imin(cb + 1, IR - 1)];
                acc = __builtin_amdgcn_wmma_f32_16x16x4_f32(
                          false, a, false, bfrag[q], (short)0, acc, false, false);
            }
            const int col   = bc * 16 + nlo;
            const int rbase = br * 16 + 8 * (lane >> 4);
            #pragma unroll
            for (int v = 0; v < 8; ++v)
                s_hb[(rbase + v) * HPITCH + col] = acc[v];
        }
    }
    __syncthreads();

    // ---- Stage C: vertical gaussian -> blurred (36x36), zero outside image ---
    #pragma unroll
    for (int k = 0; k < 6; ++k) {
        const int i = tid + k * 256;
        if (i < BR * BR) {
            const int r = i / BR;
            const int c = i - r * BR;
            const int gy = y0 - 2 + r;
            const int gx = x0 - 2 + c;
            float v = 0.0f;
            if ((unsigned)gy < (unsigned)HH && (unsigned)gx < (unsigned)WW) {
                v = g0 * s_hb[(r + 0) * HPITCH + c]
                  + g1 * s_hb[(r + 1) * HPITCH + c]
                  + g2 * s_hb[(r + 2) * HPITCH + c]
                  + g3 * s_hb[(r + 3) * HPITCH + c]
                  + g4 * s_hb[(r + 4) * HPITCH + c];
            }
            s_bl[r * BPITCH + c] = v;
        }
    }
    __syncthreads();

    // ---- Stage D: Sobel + hypot -> grad magnitude (34x34), zero outside -----
    #pragma unroll
    for (int k = 0; k < 5; ++k) {
        const int i = tid + k * 256;
        if (i < GR * GR) {
            const int r = i / GR;
            const int c = i - r * GR;
            const int gy = y0 - 1 + r;
            const int gx = x0 - 1 + c;
            float v = 0.0f;
            if ((unsigned)gy < (unsigned)HH && (unsigned)gx < (unsigned)WW) {
                const float* b = &s_bl[r * BPITCH + c];
                const float b00 = b[0],          b01 = b[1],          b02 = b[2];
                const float b10 = b[BPITCH],                          b12 = b[BPITCH + 2];
                const float b20 = b[2 * BPITCH], b21 = b[2 * BPITCH + 1], b22 = b[2 * BPITCH + 2];
                const float sx = (b00 - b02) + 2.0f * (b10 - b12) + (b20 - b22);
                const float sy = (b00 + 2.0f * b01 + b02) - (b20 + 2.0f * b21 + b22);
                v = hypotf(sx, sy);
            }
            s_gm[r * GPITCH + c] = v;
        }
    }
    __syncthreads();

    // ---- Stage E: orientation + NMS, write 32x32 outputs ---------------------
    #pragma unroll
    for (int k = 0; k < 4; ++k) {
        const int i = tid + k * 256;
        const int r = i >> 5;
        const int c = i & 31;

        // Recompute gx/gy at the output pixel from blurred LDS (cheap).
        const float* b = &s_bl[(r + 1) * BPITCH + (c + 1)];
        const float b00 = b[0],          b01 = b[1],          b02 = b[2];
        const float b10 = b[BPITCH],                          b12 = b[BPITCH + 2];
        const float b20 = b[2 * BPITCH], b21 = b[2 * BPITCH + 1], b22 = b[2 * BPITCH + 2];
        const float sx = (b00 - b02) + 2.0f * (b10 - b12) + (b20 - b22);
        const float sy = (b00 + 2.0f * b01 + b02) - (b20 + 2.0f * b21 + b22);

        // t = round((atan2_deg + 180)/45) in {0..8}; rintf = round-half-even (matches jnp.round)
        const float tdeg = atan2f(sy, sx) * 57.29577951308232f + 180.0f;
        const int t = (int)rintf(tdeg / 45.0f);
        const int ki = t & 7;   // t mod 8 (8 -> 0)

        const float* gmc = &s_gm[(r + 1) * GPITCH + (c + 1)];
        const float gm = gmc[0];
        float thin = gm;
        if (ki < 4) {
            // direction offsets: 0:(0,+1) 1:(+1,+1) 2:(+1,0) 3:(+1,-1)
            const int dr = (ki > 0) ? 1 : 0;
            const int dc = (ki < 2) ? 1 : (2 - ki);
            const int off = dr * GPITCH + dc;
            const float dpos = gm - gmc[off];    // dirs[ki]
            const float dneg = gm - gmc[-off];   // dirs[ki+4]
            if (!(fminf(dpos, dneg) > 0.0f)) thin = 0.0f;
        }
        out[base + (long long)(y0 + r) * WW + (x0 + c)] = thin;
    }
}

extern "C" void kernel_launch(void* const* d_in, const int* in_sizes, int n_in,
                              void* d_out, int out_size, void* d_ws, size_t ws_size,
                              hipStream_t stream) {
    (void)in_sizes; (void)n_in; (void)out_size; (void)d_ws; (void)ws_size;
    const float* img = (const float*)d_in[0];
    const float* wg  = (const float*)d_in[1];   // w_gauss_h: first 5 floats = taps
    float* out = (float*)d_out;
    dim3 grid(WW / TILE, HH / TILE, 16 * 3);    // x tiles, y tiles, batch*channels
    canny_fused_kernel<<<grid, 256, 0, stream>>>(img, wg, out);
}